// StateSpaceModel_88235808129391
// MI455X (gfx1250) — compile-verified
//
#include <hip/hip_runtime.h>
#include <math.h>

// Problem constants (reference: B=256, T=4096, DX=32, DU=16, DY=32)
static constexpr int kB  = 256;
static constexpr int kT  = 4096;
static constexpr int kDX = 32;
static constexpr int kDU = 16;
static constexpr int kDY = 32;
static constexpr int kCH = 32;           // time steps per block (must be even)
static constexpr int kNC = kT / kCH;     // 128 chunks
#define LOG2PI 1.8378770664093453f

typedef float v2f __attribute__((ext_vector_type(2)));
typedef float v8f __attribute__((ext_vector_type(8)));

__device__ __forceinline__ v8f wmma4(v2f a, v2f b, v8f c) {
  // D(16x16,f32) = A(16x4,f32) x B(4x16,f32) + C   -> v_wmma_f32_16x16x4_f32
  return __builtin_amdgcn_wmma_f32_16x16x4_f32(
      /*neg_a=*/false, a, /*neg_b=*/false, b,
      /*c_mod=*/(short)0, c, /*reuse_a=*/false, /*reuse_b=*/false);
}

__device__ __forceinline__ v8f load8(const float* p) {
  const float4* q = (const float4*)p;
  float4 lo = q[0], hi = q[1];
  v8f r;
  r[0] = lo.x; r[1] = lo.y; r[2] = lo.z; r[3] = lo.w;
  r[4] = hi.x; r[5] = hi.y; r[6] = hi.z; r[7] = hi.w;
  return r;
}

// ---------------------------------------------------------------------------
// Kernel 1: per-time-step broadcast vectors
//   negBuu[t][i] = -(Bu @ u_t)[i]
//   duy[t][i]    =  (Du @ u_t)[i] - ys[t][i]
// ---------------------------------------------------------------------------
__global__ void ssm_pre(const float* __restrict__ us, const float* __restrict__ ys,
                        const float* __restrict__ Bu, const float* __restrict__ Du,
                        float* __restrict__ negBuu, float* __restrict__ duy) {
  int idx = blockIdx.x * blockDim.x + threadIdx.x;    // t*32 + i
  if (idx >= kT * kDX) return;
  int t = idx >> 5, i = idx & 31;
  float s1 = 0.f, s2 = 0.f;
#pragma unroll
  for (int k = 0; k < kDU; ++k) {
    float u = us[t * kDU + k];
    s1 += Bu[i * kDU + k] * u;
    s2 += Du[i * kDU + k] * u;
  }
  negBuu[idx] = -s1;
  duy[idx] = s2 - ys[t * kDY + i];
}

// ---------------------------------------------------------------------------
// Kernel 2: main WMMA kernel. One wave32 per block.
//   grid.x = time chunk (kNC), grid.y = batch tile (kB/16)
//   e_t = [-A | I] x [x_prev; x_next] + (-Bu u)   (stays in D-layout via WMMA)
//   e_o = [ C ]    x [x_next]         + (Du u - y)
//   acc += sum_i e^2 * (-0.5 exp(-logvar))
// Time loop unrolled x2 with ping-pong fragment buffers: the x_next->x_prev
// role swap is compile-time renaming (no register copies).
// ---------------------------------------------------------------------------
__global__ __launch_bounds__(32) void ssm_main(
    const float* __restrict__ xs, const float* __restrict__ A,
    const float* __restrict__ Cm, const float* __restrict__ log_q,
    const float* __restrict__ log_r, const float* __restrict__ negBuu,
    const float* __restrict__ duy, float* __restrict__ partial) {
  const int lane = threadIdx.x;
  const int half = lane >> 4;        // 0: lanes 0-15, 1: lanes 16-31
  const int lrow = lane & 15;
  const int koff = half * 2;         // K offset within a 4-wide fragment
  const int tc = blockIdx.x;
  const int bt = blockIdx.y;
  const int t0 = tc * kCH;

  // A-operand fragments (16x4, f32): lane<16 holds (M=lrow, K=koff..koff+1).
  v2f afr[2][8];   // -A, rows tile*16+lrow, K step s covers cols 4s..4s+3
  v2f cfr[2][8];   //  C, same layout
  v2f ifr[4];      //  I block fragments (identical for both row tiles)
#pragma unroll
  for (int tile = 0; tile < 2; ++tile) {
    int row = tile * 16 + lrow;
#pragma unroll
    for (int s = 0; s < 8; ++s) {
      int k = 4 * s + koff;
      afr[tile][s] = (v2f){-A[row * 32 + k], -A[row * 32 + k + 1]};
      cfr[tile][s] = (v2f){ Cm[row * 32 + k], Cm[row * 32 + k + 1]};
    }
  }
#pragma unroll
  for (int s = 0; s < 4; ++s) {
    int j = 4 * s + koff;
    ifr[s] = (v2f){(lrow == j) ? 1.0f : 0.0f, (lrow == j + 1) ? 1.0f : 0.0f};
  }

  // Per-lane D-layout weights: -0.5 * exp(-logvar[row]), row = tile*16+half*8+r
  float wq[2][8], wr[2][8];
#pragma unroll
  for (int tile = 0; tile < 2; ++tile)
#pragma unroll
    for (int r = 0; r < 8; ++r) {
      int row = tile * 16 + half * 8 + r;
      wq[tile][r] = -0.5f * __expf(-log_q[row]);
      wr[tile][r] = -0.5f * __expf(-log_r[row]);
    }

  // B-operand fragments (4x16): lane holds (K=4s+koff..+1, N=lrow).
  // Column n of the batch tile = batch bt*16 + lrow.
  const size_t xb = (size_t)(bt * 16 + lrow) * (size_t)(kT + 1) * kDX;

  float acc = 0.0f;
  v2f b0[8], b1[8];   // ping-pong x fragment buffers

  // One SSM step: consumes xp, loads x_{t+1} into xn, accumulates into acc.
  auto step = [&](int t, v2f (&xp)[8], v2f (&xn)[8]) {
    // Issue loads first: x_next fragments + broadcast C-operands.
    const float2* p = (const float2*)(xs + xb + (size_t)(t + 1) * kDX);
#pragma unroll
    for (int s = 0; s < 8; ++s) { float2 v = p[2 * s + half]; xn[s] = (v2f){v.x, v.y}; }
    v8f dt0 = load8(negBuu + t * kDX + half * 8);
    v8f dt1 = load8(negBuu + t * kDX + 16 + half * 8);
    v8f do0 = load8(duy + t * kDY + half * 8);
    v8f do1 = load8(duy + t * kDY + 16 + half * 8);

    // Phase 1: -A x_prev (independent of the in-flight xn loads).
#pragma unroll
    for (int s = 0; s < 8; ++s) {
      dt0 = wmma4(afr[0][s], xp[s], dt0);
      dt1 = wmma4(afr[1][s], xp[s], dt1);
    }
    // Phase 2: xn-dependent chains (obs + identity add of x_next).
#pragma unroll
    for (int s = 0; s < 8; ++s) {
      do0 = wmma4(cfr[0][s], xn[s], do0);
      do1 = wmma4(cfr[1][s], xn[s], do1);
    }
#pragma unroll
    for (int s = 0; s < 4; ++s) {
      dt0 = wmma4(ifr[s], xn[s], dt0);
      dt1 = wmma4(ifr[s], xn[4 + s], dt1);
    }

#pragma unroll
    for (int r = 0; r < 8; ++r)
      acc += dt0[r] * dt0[r] * wq[0][r] + dt1[r] * dt1[r] * wq[1][r]
           + do0[r] * do0[r] * wr[0][r] + do1[r] * do1[r] * wr[1][r];
  };

  // Preload x_{t0} into b0.
  {
    const float2* p = (const float2*)(xs + xb + (size_t)t0 * kDX);
#pragma unroll
    for (int s = 0; s < 8; ++s) { float2 v = p[2 * s + half]; b0[s] = (v2f){v.x, v.y}; }
  }

  for (int i = 0; i < kCH; i += 2) {
    step(t0 + i,     b0, b1);   // xp = b0, xn -> b1
    step(t0 + i + 1, b1, b0);   // xp = b1, xn -> b0 (pure renaming, no movs)
  }

  // Lanes n and n+16 hold the two row-halves of batch column n: combine.
  __shared__ float red[32];
  red[lane] = acc;
  __syncthreads();
  if (lane < 16)
    partial[(size_t)tc * kB + bt * 16 + lane] = red[lane] + red[lane + 16];
}

// ---------------------------------------------------------------------------
// Kernel 3: reduction — lp0, analytic constants, chunk partials (deterministic)
// ---------------------------------------------------------------------------
__global__ void ssm_reduce(const float* __restrict__ xs, const float* __restrict__ log_q,
                           const float* __restrict__ log_r, const float* __restrict__ mu0,
                           const float* __restrict__ log_s0,
                           const float* __restrict__ partial, float* __restrict__ out) {
  int b = blockIdx.x * blockDim.x + threadIdx.x;
  if (b >= kB) return;
  float lp0 = 0.f, cst = 0.f;
#pragma unroll 4
  for (int i = 0; i < kDX; ++i) {
    float d = xs[(size_t)b * (kT + 1) * kDX + i] - mu0[i];
    lp0 += d * d * __expf(-log_s0[i]) + log_s0[i] + LOG2PI;
    cst += (log_q[i] + LOG2PI) + (log_r[i] + LOG2PI);
  }
  float s = 0.f;
  for (int c = 0; c < kNC; ++c) s += partial[(size_t)c * kB + b];
  out[b] = -0.5f * lp0 + s - 0.5f * (float)kT * cst;
}

// ---------------------------------------------------------------------------
extern "C" void kernel_launch(void* const* d_in, const int* in_sizes, int n_in,
                              void* d_out, int out_size, void* d_ws, size_t ws_size,
                              hipStream_t stream) {
  const float* xs     = (const float*)d_in[0];
  const float* us     = (const float*)d_in[1];
  const float* ys     = (const float*)d_in[2];
  const float* A      = (const float*)d_in[3];
  const float* Bu     = (const float*)d_in[4];
  const float* log_q  = (const float*)d_in[5];
  const float* Cm     = (const float*)d_in[6];
  const float* Du     = (const float*)d_in[7];
  const float* log_r  = (const float*)d_in[8];
  const float* mu0    = (const float*)d_in[9];
  const float* log_s0 = (const float*)d_in[10];
  float* out = (float*)d_out;

  // Workspace: negBuu (T*32) | duy (T*32) | partial (NC*B)  ~= 1.2 MB
  float* negBuu  = (float*)d_ws;
  float* duy     = negBuu + (size_t)kT * kDX;
  float* partial = duy + (size_t)kT * kDX;

  ssm_pre<<<(kT * kDX + 255) / 256, 256, 0, stream>>>(us, ys, Bu, Du, negBuu, duy);

  dim3 grid(kNC, kB / 16);
  ssm_main<<<grid, 32, 0, stream>>>(xs, A, Cm, log_q, log_r, negBuu, duy, partial);

  ssm_reduce<<<1, kB, 0, stream>>>(xs, log_q, log_r, mu0, log_s0, partial, out);
}